// SASA_26439818674369
// MI455X (gfx1250) — compile-verified
//
#include <hip/hip_runtime.h>
#include <math.h>

// ---------------------------------------------------------------------------
// SASA forward, algebraically collapsed.
//
//   s[b,i,j]        = sum_c x_padded[b,i,j,c]                (channel sum)
//   wqsum[c]        = sum_o Wq[o,c]        (and per-tap wksum / wvsum)
//   qv              = s[b,i+2,j+2] * wqsum[c]
//   logit[t]        = (s_tap[t]*wksum[t,c] + pos[t,c]) * qv
//   out[b,i,j,c]    = sum_t softmax_t(logit)[t] * s_tap[t] * wvsum[t,c]
//
// Column sums are a genuine (ones-vector x matrix) product -> done with
// V_WMMA_F32_16X16X4_F32 batched over the 99 weight matrices.
// ---------------------------------------------------------------------------

typedef __attribute__((ext_vector_type(2))) float v2f;
typedef __attribute__((ext_vector_type(8))) float v8f;

#define KS    7
#define TAPS  49      // KS*KS
#define CC    128     // COUT == CIN
#define HH    56
#define WW    56
#define BB    2
#define SP    65      // padded s grid: indices u=i+p (i<56, p<7) and u=i+5 all < 65
#define NMAT  99      // 1 (Wq) + 49 (Wk) + 49 (Wv)

// ---- Kernel 1: zero-framed channel-sum grid -------------------------------
// spad[b][u][v] = sum_c x[b, u-6, v-6, c] when (u-6,v-6) in [0,56)^2 else 0.
// (s lives at grid index (i+3); the extra +3 absorbs the 7-tap offset p-3.)
__global__ void sasa_spad(const float* __restrict__ x, float* __restrict__ spad) {
    int idx = blockIdx.x * blockDim.x + threadIdx.x;
    if (idx >= BB * SP * SP) return;
    int v = idx % SP;
    int u = (idx / SP) % SP;
    int b = idx / (SP * SP);
    float sum = 0.0f;
    if (u >= 6 && u < 6 + HH && v >= 6 && v < 6 + WW) {
        const float4* p =
            (const float4*)(x + ((size_t)((b * HH + (u - 6)) * WW + (v - 6))) * CC);
        #pragma unroll 8
        for (int c4 = 0; c4 < CC / 4; ++c4) {
            float4 t = p[c4];
            sum += (t.x + t.y) + (t.z + t.w);
        }
    }
    spad[idx] = sum;
}

// ---- Kernel 2: batched column sums via WMMA f32 16x16x4 -------------------
// One wave per (matrix, 16-column tile): D = ones(16x4) * W_chunk(4x16) + D.
// A layout (f32 16x4): lanes 0-15 -> {K0,K1}, lanes 16-31 -> {K2,K3}; all 1.0.
// B layout (f32 4x16): lane n=lane&15 is column N; vgpr0/1 = rows
//   (K0,K1) for lanes 0-15 and (K2,K3) for lanes 16-31.
// Since A is all ones, every D row equals the column sum -> read acc[0],
// lanes 0-15 (M=0 row, N=lane).
__global__ void sasa_colsum(const float* __restrict__ Wq,
                            const float* __restrict__ Wk,
                            const float* __restrict__ Wv,
                            float* __restrict__ wsum) {
    int wid = blockIdx.x;          // 0..791
    int m   = wid >> 3;            // matrix id 0..98
    int ct  = wid & 7;             // 16-column tile 0..7
    const float* W =
        (m == 0) ? Wq
                 : (m <= TAPS ? Wk + (size_t)(m - 1) * CC * CC
                              : Wv + (size_t)(m - 1 - TAPS) * CC * CC);
    int lane = threadIdx.x;        // 0..31 (wave32)
    int n    = lane & 15;
    int khi  = (lane >> 4) << 1;   // 0 or 2

    v2f a;  a.x = 1.0f; a.y = 1.0f;
    v8f acc = {};

    const float* col = W + ct * 16 + n;
    for (int k0 = 0; k0 < CC; k0 += 4) {
        v2f bm;
        bm.x = col[(size_t)(k0 + khi + 0) * CC];
        bm.y = col[(size_t)(k0 + khi + 1) * CC];
        acc = __builtin_amdgcn_wmma_f32_16x16x4_f32(
            /*neg_a=*/false, a, /*neg_b=*/false, bm,
            /*c_mod=*/(short)0, acc, /*reuse_a=*/false, /*reuse_b=*/false);
    }
    if (lane < 16) wsum[(size_t)m * CC + ct * 16 + lane] = acc[0];
}

// ---- Kernel 3: per-pixel 49-tap softmax attention -------------------------
// Block = one output pixel, 128 threads = one channel each (4 wave32).
// The 49 tap scalars are shared across channels -> staged once in LDS.
__global__ void sasa_attn(const float* __restrict__ spad,
                          const float* __restrict__ wsum,
                          const float* __restrict__ pos,
                          float* __restrict__ out) {
    __shared__ float s_sh[TAPS];
    int pix = blockIdx.x;                 // b-major: ((b*56+i)*56+j)
    int b   = pix / (HH * WW);
    int ij  = pix % (HH * WW);
    int i   = ij / WW;
    int j   = ij % WW;

    int t = threadIdx.x;
    if (t < TAPS) {
        int p = t / KS, q = t % KS;       // tap scalar s[b, i+p-3, j+q-3]
        s_sh[t] = spad[((size_t)b * SP + (i + p)) * SP + (j + q)];
    }
    __syncthreads();

    int c = threadIdx.x;                  // channel
    const float* wqs = wsum;              // [128]
    const float* wks = wsum + CC;         // [49*128]
    const float* wvs = wsum + CC + (size_t)TAPS * CC;

    // sq = s[b, i+2, j+2] == tap (p=5,q=5) of the staged window
    float qv = s_sh[5 * KS + 5] * wqs[c];

    float l[TAPS];
    float mx = -3.402823466e38f;
    #pragma unroll
    for (int tt = 0; tt < TAPS; ++tt) {
        float lv = fmaf(s_sh[tt], wks[(size_t)tt * CC + c],
                        pos[(size_t)tt * CC + c]) * qv;
        l[tt] = lv;
        mx = fmaxf(mx, lv);
    }
    float den = 0.0f, num = 0.0f;
    #pragma unroll
    for (int tt = 0; tt < TAPS; ++tt) {
        float e = __expf(l[tt] - mx);
        den += e;
        num = fmaf(e * s_sh[tt], wvs[(size_t)tt * CC + c], num);
    }
    out[(size_t)pix * CC + c] = num / den;
}

// ---------------------------------------------------------------------------
extern "C" void kernel_launch(void* const* d_in, const int* in_sizes, int n_in,
                              void* d_out, int out_size, void* d_ws, size_t ws_size,
                              hipStream_t stream) {
    const float* x   = (const float*)d_in[0];
    const float* Wq  = (const float*)d_in[1];
    const float* Wk  = (const float*)d_in[2];
    const float* Wv  = (const float*)d_in[3];
    const float* pos = (const float*)d_in[4];
    float* out = (float*)d_out;

    // workspace layout: [ spad: 2*65*65 | wsum: 99*128 ] floats (~84 KB)
    float* spad = (float*)d_ws;
    float* wsum = spad + BB * SP * SP;

    int nspad = BB * SP * SP;
    sasa_spad  <<<(nspad + 127) / 128, 128, 0, stream>>>(x, spad);
    sasa_colsum<<<NMAT * 8,            32, 0, stream>>>(Wq, Wk, Wv, wsum);
    sasa_attn  <<<BB * HH * WW,        CC, 0, stream>>>(spad, wsum, pos, out);
}